// Sinkhorn_76416058130512
// MI455X (gfx1250) — compile-verified
//
#include <hip/hip_runtime.h>
#include <stdint.h>

// Sinkhorn [B,512,512], 10 iters, probability-domain reformulation:
//   rinv[i] = 1 / sum_j exp(x[i,j]) * cinv[j]
//   cinv[j] = 1 / sum_i exp(x[i,j]) * rinv[i]
//   out[i,j] = exp(x[i,j]) * rinv[i] * cinv[j]
// One workgroup per matrix; 32-row tiles double-buffered through LDS via
// global_load_async_to_lds_b128 (ASYNCcnt). Row reduce: v_exp + FMA + wave32
// butterfly. Col reduce: V_WMMA_F32_16X16X4_F32 accumulation (A = rinv
// replicated over M, B = 4x16 chunk of exp-tile, C = per-wave 16-col f32
// accumulator held across all tiles of an iteration).

#define NN      512
#define TILE_R  32
#define NT      (NN / TILE_R)     // 16 tiles per matrix
#define BLOCK   1024              // 32 waves (wave32)
#define ITERS   10

typedef float v2f __attribute__((ext_vector_type(2)));
typedef float v8f __attribute__((ext_vector_type(8)));

#if __has_builtin(__builtin_amdgcn_s_wait_asynccnt)
#define WAIT_ASYNC0() __builtin_amdgcn_s_wait_asynccnt(0)
#else
#define WAIT_ASYNC0() asm volatile("s_wait_asynccnt 0" ::: "memory")
#endif

#if __has_builtin(__builtin_amdgcn_wmma_f32_16x16x4_f32)
#define HAVE_WMMA_F32X4 1
#else
#define HAVE_WMMA_F32X4 0
#endif

// Stage one 32x512 f32 tile (64KB) from global into LDS.
// 1024 threads x 4 issues x 16B = 64KB; per wave: 4 async ops on ASYNCcnt.
__device__ __forceinline__ void async_tile_load(const float* gsrc, float* ldst, int tid) {
#pragma unroll
    for (int k = 0; k < 4; ++k) {
        const int e = tid * 4 + k * 4096;                       // float index, 16B aligned
        unsigned           lo = (unsigned)(uintptr_t)(ldst + e); // LDS byte offset
        unsigned long long ga = (unsigned long long)(uintptr_t)(gsrc + e);
        asm volatile("global_load_async_to_lds_b128 %0, %1, off"
                     :: "v"(lo), "v"(ga) : "memory");
    }
}

extern "C" __global__ __launch_bounds__(BLOCK, 1)
void sinkhorn_cdna5_kernel(const float* __restrict__ la, float* __restrict__ out) {
    __shared__ float sT[2][TILE_R * NN];   // double-buffered tile (2 x 64KB)
    __shared__ float s_rinv[NN];           // exp(-r[i]) = 1/rowsum
    __shared__ float s_cinv[NN];           // exp(-c[j]) = 1/colsum
#if !HAVE_WMMA_F32X4
    __shared__ float s_cpart[NN];          // upper-half column partials (fallback)
#endif

    const int tid  = threadIdx.x;
    const int wave = tid >> 5;
    const int lane = tid & 31;

    const float* gmat = la  + (size_t)blockIdx.x * NN * NN;
    float*       gout = out + (size_t)blockIdx.x * NN * NN;

    if (tid < NN) s_cinv[tid] = 1.0f;      // c = 0  ->  exp(-c) = 1

    async_tile_load(gmat, &sT[0][0], tid); // prologue: tile 0 in flight

#if HAVE_WMMA_F32X4
    const int hi = lane >> 4;              // lane half (K striping of A/B operands)
    const int nc = lane & 15;              // N index within the wave's column group
    const int cb = wave * 16;              // this wave owns columns cb..cb+15
#else
    const int jcol = tid & (NN - 1);
    const int half = tid >> 9;
#endif

    for (int it = 0; it <= ITERS; ++it) {
#if HAVE_WMMA_F32X4
        v8f cC = {};                       // column-sum accumulator (16x16 f32 C/D)
#else
        float cacc = 0.0f;
#endif
        for (int t = 0; t < NT; ++t) {
            const int buf = t & 1;
            WAIT_ASYNC0();                 // my lanes' tile-t data landed
            __syncthreads();               // everyone's landed; prev tile compute done
            // Issue the next tile into the other buffer (overlaps this tile's compute).
            if (!(it == ITERS && t == NT - 1)) {
                const int nt = (t + 1) & (NT - 1);   // wraps to tile 0 of next sweep
                async_tile_load(gmat + (size_t)nt * TILE_R * NN, &sT[nt & 1][0], tid);
            }

            if (it < ITERS) {
                // ---- row pass: wave w owns row t*32+w; converts tile to exp-domain
                {
                    float* rowp = &sT[buf][wave * NN];
                    float  acc  = 0.0f;
#pragma unroll
                    for (int k = 0; k < NN / 32; ++k) {
                        const int   j = lane + k * 32;
                        const float E = __expf(rowp[j]);   // one v_exp per elem per iter
                        rowp[j] = E;                       // reused by col pass below
                        acc = fmaf(E, s_cinv[j], acc);
                    }
#pragma unroll
                    for (int off = 16; off > 0; off >>= 1)
                        acc += __shfl_xor(acc, off, 32);   // wave32 butterfly reduce
                    if (lane == 0) s_rinv[t * TILE_R + wave] = 1.0f / acc;
                }
                __syncthreads();           // E-tile + rinv visible
                // ---- col pass
#if HAVE_WMMA_F32X4
                {
                    const int R0 = t * TILE_R;
#pragma unroll
                    for (int kb = 0; kb < TILE_R / 4; ++kb) {
                        const int r0  = 4 * kb;
                        const int ka0 = r0 + (hi ? 2 : 0); // K rows in VGPR0 (lanes 0-15 / 16-31)
                        const int ka1 = r0 + (hi ? 3 : 1); // K rows in VGPR1
                        v2f A, B;
                        A.x = s_rinv[R0 + ka0];            // A[m,k]=rinv[k], replicated over M
                        A.y = s_rinv[R0 + ka1];
                        B.x = sT[buf][ka0 * NN + cb + nc]; // B[k,n]=E[k, cb+n]
                        B.y = sT[buf][ka1 * NN + cb + nc];
                        cC = __builtin_amdgcn_wmma_f32_16x16x4_f32(
                                false, A, false, B, (short)0, cC, false, false);
                    }
                }
#else
                {
#pragma unroll
                    for (int k = 0; k < TILE_R / 2; ++k) {
                        const int lr = half * (TILE_R / 2) + k;
                        cacc = fmaf(sT[buf][lr * NN + jcol],
                                    s_rinv[t * TILE_R + lr], cacc);
                    }
                }
#endif
            } else {
                // ---- final sweep: out = exp(x) * rinv[i] * cinv[j], coalesced stores
#pragma unroll
                for (int k = 0; k < (TILE_R * NN) / BLOCK; ++k) {
                    const int idx = tid + k * BLOCK;
                    const int lr  = idx >> 9;
                    const int j   = idx & (NN - 1);
                    const float v = __expf(sT[buf][idx]) *
                                    s_rinv[t * TILE_R + lr] * s_cinv[j];
                    gout[(size_t)(t * TILE_R + lr) * NN + j] = v;
                }
            }
        }
        if (it < ITERS) {
#if HAVE_WMMA_F32X4
            // All M rows of D are identical (A replicated); row M=0 lives in
            // cC[0] of lanes 0..15 (C/D layout): lane n holds colsum[cb+n].
            if (lane < 16) s_cinv[cb + lane] = 1.0f / cC[0];
#else
            if (half == 1) s_cpart[jcol] = cacc;
            __syncthreads();
            if (half == 0) s_cinv[jcol] = 1.0f / (cacc + s_cpart[jcol]);
#endif
            // next sweep's first ready-sync fences these writes before they're read
        }
    }
}

extern "C" void kernel_launch(void* const* d_in, const int* in_sizes, int n_in,
                              void* d_out, int out_size, void* d_ws, size_t ws_size,
                              hipStream_t stream) {
    const float* la  = (const float*)d_in[0];
    float*       out = (float*)d_out;
    const int batches = in_sizes[0] / (NN * NN);   // 256
    sinkhorn_cdna5_kernel<<<dim3(batches), dim3(BLOCK), 0, stream>>>(la, out);
}